// ProbabilityDropout_10471130268488
// MI455X (gfx1250) — compile-verified
//
#include <hip/hip_runtime.h>

typedef float v2f __attribute__((ext_vector_type(2)));
typedef float v8f __attribute__((ext_vector_type(8)));

#define BDIM   1024
#define DLAT   512
#define NBINS  32768
#define ROWLEN 8192   // D*M
#define NWAVE  (BDIM/32)

// LDS layout (dynamic):
//   [0,      32768)  : s_eps   8192 f32
//   [32768,  163840) : s_hist  32768 u32  (reused as E[] f32 after counts consumed)
//   [163840, 165888) : s_mean  512 f32
//   [165888, 167936) : s_scale 512 f32
//   [167936, 168064) : s_redA  32 f32
//   [168064, 168192) : s_redB  32 f32
//   [168192, 168208) : s_scal  4 f32 (pmin, pmax, maxc, sumexp)
#define SMEM_BYTES 168208

__global__ __launch_bounds__(BDIM)
void vae_hist_softmax_kernel(const float* __restrict__ z_mean,
                             const float* __restrict__ z_var,
                             const float* __restrict__ x,
                             const float* __restrict__ epsilon,
                             float* __restrict__ out)
{
    extern __shared__ char smem[];
    float*    s_eps   = (float*)(smem);
    unsigned* s_hist  = (unsigned*)(smem + 32768);
    float*    s_E     = (float*)(smem + 32768);     // alias of s_hist
    float*    s_mean  = (float*)(smem + 163840);
    float*    s_scale = (float*)(smem + 165888);
    float*    s_redA  = (float*)(smem + 167936);
    float*    s_redB  = (float*)(smem + 168064);
    float*    s_scal  = (float*)(smem + 168192);

    const int tid  = threadIdx.x;
    const int b    = blockIdx.x;
    const int wave = tid >> 5;
    const int lane = tid & 31;

    // ---- Phase 0: async-copy epsilon row (32 KB) global -> LDS (ASYNCcnt path)
    {
        const float* gsrc = epsilon + (size_t)b * ROWLEN;
        unsigned ldsAddr = (unsigned)(uintptr_t)(&s_eps[0]) + (unsigned)(tid * 32);
        unsigned gOff    = (unsigned)(tid * 32);
        unsigned long long gbase = (unsigned long long)(uintptr_t)gsrc;
        asm volatile(
            "global_load_async_to_lds_b128 %0, %1, %2 offset:0\n\t"
            "global_load_async_to_lds_b128 %0, %1, %2 offset:16"
            :: "v"(ldsAddr), "v"(gOff), "s"(gbase) : "memory");
    }

    // Overlap with the async copy: zero histogram, precompute mean / exp(0.5*var)
    #pragma unroll
    for (int i = 0; i < NBINS / BDIM; ++i)
        s_hist[tid + i * BDIM] = 0u;

    if (tid < DLAT) {
        s_mean[tid]  = z_mean[(size_t)b * DLAT + tid];
        s_scale[tid] = expf(0.5f * z_var[(size_t)b * DLAT + tid]);
    }

    asm volatile("s_wait_asynccnt 0" ::: "memory");
    __syncthreads();

    // ---- Phase 1: z into registers, row min/max reduction
    float zreg[ROWLEN / BDIM];
    float lmin = 3.402823466e38f, lmax = -3.402823466e38f;
    #pragma unroll
    for (int i = 0; i < ROWLEN / BDIM; ++i) {
        int e = tid + i * BDIM;
        int d = e & (DLAT - 1);
        // keep mul+add unfused to match reference bin boundaries
        float z = __fadd_rn(s_mean[d], __fmul_rn(s_scale[d], s_eps[e]));
        zreg[i] = z;
        lmin = fminf(lmin, z);
        lmax = fmaxf(lmax, z);
    }
    #pragma unroll
    for (int m = 16; m; m >>= 1) {
        lmin = fminf(lmin, __shfl_xor(lmin, m, 32));
        lmax = fmaxf(lmax, __shfl_xor(lmax, m, 32));
    }
    if (lane == 0) { s_redA[wave] = lmin; s_redB[wave] = lmax; }
    __syncthreads();
    if (wave == 0) {
        float a = s_redA[lane], c = s_redB[lane];
        #pragma unroll
        for (int m = 16; m; m >>= 1) {
            a = fminf(a, __shfl_xor(a, m, 32));
            c = fmaxf(c, __shfl_xor(c, m, 32));
        }
        if (lane == 0) { s_scal[0] = a; s_scal[1] = c; }
    }
    __syncthreads();
    const float pmin  = s_scal[0];
    const float width = fmaxf(s_scal[1] - pmin, 1e-12f);

    // ---- Phase 2: histogram via LDS atomics
    #pragma unroll
    for (int i = 0; i < ROWLEN / BDIM; ++i) {
        float t = (zreg[i] - pmin) / width;     // same op order as reference
        int bin = (int)floorf(t * 32768.0f);
        bin = min(max(bin, 0), NBINS - 1);
        atomicAdd(&s_hist[bin], 1u);
    }
    __syncthreads();

    // ---- Phase 3: max count
    int lmc = 0;
    #pragma unroll
    for (int i = 0; i < NBINS / BDIM; ++i)
        lmc = max(lmc, (int)s_hist[tid + i * BDIM]);
    #pragma unroll
    for (int m = 16; m; m >>= 1)
        lmc = max(lmc, __shfl_xor(lmc, m, 32));
    if (lane == 0) s_redA[wave] = (float)lmc;
    __syncthreads();
    if (wave == 0) {
        float a = s_redA[lane];
        #pragma unroll
        for (int m = 16; m; m >>= 1) a = fmaxf(a, __shfl_xor(a, m, 32));
        if (lane == 0) s_scal[2] = a;
    }
    __syncthreads();
    const float maxc = s_scal[2];

    // ---- Phase 4: E[k] = exp(c[k]-maxc); denominator via WMMA f32 16x16x4, B = ones
    // Each wave reduces its exclusive 1024-bin chunk with 16 chained WMMAs.
    // D = A*ones + C => each D row replicates rowsum(A); column-sum of D = chunk total.
    v8f acc = {};
    const v2f bones = {1.0f, 1.0f};
    const int base = wave * 1024;
    #pragma unroll
    for (int t = 0; t < 16; ++t) {
        int k0 = base + t * 64 + lane * 2;
        float e0 = expf((float)s_hist[k0]     - maxc);
        float e1 = expf((float)s_hist[k0 + 1] - maxc);
        s_E[k0] = e0;            // overwrite counts (this wave owns the chunk)
        s_E[k0 + 1] = e1;
        v2f a = {e0, e1};
        acc = __builtin_amdgcn_wmma_f32_16x16x4_f32(
            /*neg_a=*/false, a, /*neg_b=*/false, bones,
            /*c_mod=*/(short)0, acc, /*reuse_a=*/false, /*reuse_b=*/false);
    }
    float s = acc[0] + acc[1] + acc[2] + acc[3] + acc[4] + acc[5] + acc[6] + acc[7];
    s += __shfl_xor(s, 16, 32);           // lane L + lane L^16 = full column sum = chunk total
    if (lane == 0) s_redA[wave] = s;
    __syncthreads();
    if (wave == 0) {
        float a = s_redA[lane];
        #pragma unroll
        for (int m = 16; m; m >>= 1) a += __shfl_xor(a, m, 32);
        if (lane == 0) s_scal[3] = a;
    }
    __syncthreads();
    const float sumexp = s_scal[3];

    // ---- Phase 5: out = x * softmax(hist) * M
    const size_t rowbase = (size_t)b * NBINS;
    #pragma unroll
    for (int i = 0; i < NBINS / BDIM; ++i) {
        int k = tid + i * BDIM;
        float prob = s_E[k] / sumexp;
        out[rowbase + k] = x[rowbase + k] * prob * 16.0f;
    }
}

extern "C" void kernel_launch(void* const* d_in, const int* in_sizes, int n_in,
                              void* d_out, int out_size, void* d_ws, size_t ws_size,
                              hipStream_t stream) {
    (void)in_sizes; (void)n_in; (void)d_ws; (void)ws_size; (void)out_size;
    const float* z_mean  = (const float*)d_in[0];
    const float* z_var   = (const float*)d_in[1];
    const float* x       = (const float*)d_in[2];
    const float* epsilon = (const float*)d_in[3];
    float* out = (float*)d_out;

    dim3 grid(2048);
    dim3 block(BDIM);
    vae_hist_softmax_kernel<<<grid, block, SMEM_BYTES, stream>>>(
        z_mean, z_var, x, epsilon, out);
}